// SinkhornDistance_53206054863637
// MI455X (gfx1250) — compile-verified
//
#include <hip/hip_runtime.h>
#include <math.h>

// ---- problem constants (match reference) ----
#define NB 4
#define P  2048
#define D  64
#define EPSV    0.1f
#define INV_EPS 10.0f
#define MAX_ITER 50
#define THRESHV  1e-9f

typedef __attribute__((ext_vector_type(2))) float v2f;
typedef __attribute__((ext_vector_type(8))) float v8f;

__device__ __forceinline__ float log_mu() {
  // log(1/P + 1e-8); compiler constant-folds
  return logf(1.0f / (float)P + 1e-8f);
}

// online logsumexp accumulate: (m,s) <- merge with value
__device__ __forceinline__ void lse_add(float& m, float& s, float val) {
  float mx = fmaxf(m, val);
  s = s * __expf(m - mx) + __expf(val - mx);
  m = mx;
}
// merge two (m,s) states
__device__ __forceinline__ void lse_merge(float& m, float& s, float m2, float s2) {
  float mx = fmaxf(m, m2);
  s = s * __expf(m - mx) + s2 * __expf(m2 - mx);
  m = mx;
}

// ---------------- init ----------------
__global__ void init_uv(float* u, float* v, int* flag) {
  int t = blockIdx.x * blockDim.x + threadIdx.x;
  if (t < NB * P) { u[t] = 0.0f; v[t] = 0.0f; }
  if (t == 0) *flag = 0;
}

// ---------------- row norms ----------------
// one wave per row; rows [0, NB*P) -> x, rows [NB*P, 2*NB*P) -> y
__global__ void row_norms(const float* __restrict__ x, const float* __restrict__ y,
                          float* __restrict__ xs, float* __restrict__ ys) {
  int wave = (blockIdx.x * blockDim.x + threadIdx.x) >> 5;
  int lane = threadIdx.x & 31;
  const float* src = x;
  float* dst = xs;
  int r = wave;
  if (r >= NB * P) { r -= NB * P; src = y; dst = ys; }
  const float2* row2 = (const float2*)(src + (size_t)r * D);
  float2 a = row2[lane];                      // one b64 per lane covers the row
  float ss = a.x * a.x + a.y * a.y;
  #pragma unroll
  for (int off = 16; off; off >>= 1) ss += __shfl_xor(ss, off, 32);
  if (lane == 0) dst[r] = ss;
}

// ---------------- cost matrix via fp32 WMMA ----------------
// C[n,i,j] = |x_i|^2 + |y_j|^2 - 2 * <x_i, y_j>
// one wave per 16x16 tile; K=64 via 16x V_WMMA_F32_16X16X4_F32
__global__ void cost_wmma(const float* __restrict__ x, const float* __restrict__ y,
                          const float* __restrict__ xs, const float* __restrict__ ys,
                          float* __restrict__ Cout) {
  int wave = (blockIdx.x * blockDim.x + threadIdx.x) >> 5;
  int lane = threadIdx.x & 31;
  int n   = wave >> 14;        // / (128*128)
  int rem = wave & 16383;
  int ti  = rem >> 7;          // row tile
  int tj  = rem & 127;         // col tile

  const float* X = x + (size_t)n * P * D;
  const float* Y = y + (size_t)n * P * D;

  // A (16x4 f32): lanes 0-15 -> M=lane, VGPR0=K0 VGPR1=K1; lanes 16-31 -> M=lane-16, K2/K3
  int m  = lane & 15;
  int kb = (lane >> 4) << 1;
  const float* xrow = X + (size_t)(ti * 16 + m) * D;
  const float* yrow = Y + (size_t)(tj * 16 + m) * D; // B = Y^T : same lane pattern on Y rows

  v8f acc = {};
  #pragma unroll
  for (int k = 0; k < D; k += 4) {
    v2f a, b;
    a.x = xrow[k + kb];
    a.y = xrow[k + kb + 1];
    b.x = yrow[k + kb];
    b.y = yrow[k + kb + 1];
    acc = __builtin_amdgcn_wmma_f32_16x16x4_f32(
        /*neg_a=*/false, a, /*neg_b=*/false, b,
        /*c_mod=*/(short)0, acc, /*reuse_a=*/false, /*reuse_b=*/false);
  }

  // C/D layout: VGPR v, lanes 0-15: (M=v, N=lane); lanes 16-31: (M=v+8, N=lane-16)
  int nc   = lane & 15;
  int half = lane >> 4;
  float ysv = ys[n * P + tj * 16 + nc];
  float* Cn = Cout + (size_t)n * P * P;
  #pragma unroll
  for (int vv = 0; vv < 8; ++vv) {
    int grow = ti * 16 + vv + half * 8;
    float c = xs[n * P + grow] + ysv - 2.0f * acc[vv];
    Cn[(size_t)grow * P + tj * 16 + nc] = c;
  }
}

// ---------------- u update (row logsumexp, b128 streaming) ----------------
// u_i = eps*(log_mu - lse_j((v_j - C_ij)/eps)); one wave per row, 8 rows/block
__global__ void u_update(const float* __restrict__ C, const float* __restrict__ v,
                         float* __restrict__ u, float* __restrict__ errPartial,
                         const int* __restrict__ flag) {
  __shared__ float deltas[8];
  if (*flag) return;  // uniform, all threads exit before any barrier
  int w = threadIdx.x >> 5, lane = threadIdx.x & 31;
  int rg = blockIdx.x * 8 + w;           // global row (same batch for whole block)
  int n  = rg >> 11;
  const float4* C4 = (const float4*)(C + (size_t)rg * P);
  const float4* v4 = (const float4*)(v + n * P);

  float m = -__builtin_inff(), s = 0.0f;
  #pragma unroll 4
  for (int it = 0; it < P / 128; ++it) {   // 16 iterations, float4 per lane
    int q = lane + it * 32;
    float4 cv = C4[q];
    float4 vv = v4[q];
    lse_add(m, s, (vv.x - cv.x) * INV_EPS);
    lse_add(m, s, (vv.y - cv.y) * INV_EPS);
    lse_add(m, s, (vv.z - cv.z) * INV_EPS);
    lse_add(m, s, (vv.w - cv.w) * INV_EPS);
  }
  #pragma unroll
  for (int off = 16; off; off >>= 1) {
    float m2 = __shfl_xor(m, off, 32);
    float s2 = __shfl_xor(s, off, 32);
    lse_merge(m, s, m2, s2);
  }
  if (lane == 0) {
    float lse = m + __logf(s);
    float un  = EPSV * (log_mu() - lse);
    deltas[w] = fabsf(un - u[rg]);
    u[rg] = un;
  }
  __syncthreads();
  if (threadIdx.x == 0) {
    float t = 0.0f;
    #pragma unroll
    for (int i = 0; i < 8; ++i) t += deltas[i];
    errPartial[blockIdx.x] = t;
  }
}

// ---------------- v update (column logsumexp, b128 streaming) ----------------
// v_j = eps*(log_nu - lse_i((u_i - C_ij)/eps))
// block = 16 column-quads (64 cols) x 16 row-chunks; float4 per thread per row
__global__ void v_update(const float* __restrict__ C, const float* __restrict__ u,
                         float* __restrict__ v, const int* __restrict__ flag) {
  __shared__ float shm[16][64];
  __shared__ float shs[16][64];
  if (*flag) return;
  int n  = blockIdx.x >> 5;        // 32 column-groups (64 cols each) per batch
  int jg = blockIdx.x & 31;
  int cq = threadIdx.x & 15;       // column quad within group
  int rc = threadIdx.x >> 4;       // row chunk 0..15
  const float4* Cn4 = (const float4*)(C + (size_t)n * P * P);
  const float* ub   = u + n * P;
  int qcol = jg * 16 + cq;         // float4 column index

  float m0 = -__builtin_inff(), m1 = m0, m2 = m0, m3 = m0;
  float s0 = 0.f, s1 = 0.f, s2 = 0.f, s3 = 0.f;
  for (int i = rc; i < P; i += 16) {
    float ui = ub[i];
    float4 c = Cn4[(size_t)i * (P / 4) + qcol];
    lse_add(m0, s0, (ui - c.x) * INV_EPS);
    lse_add(m1, s1, (ui - c.y) * INV_EPS);
    lse_add(m2, s2, (ui - c.z) * INV_EPS);
    lse_add(m3, s3, (ui - c.w) * INV_EPS);
  }
  shm[rc][cq * 4 + 0] = m0; shs[rc][cq * 4 + 0] = s0;
  shm[rc][cq * 4 + 1] = m1; shs[rc][cq * 4 + 1] = s1;
  shm[rc][cq * 4 + 2] = m2; shs[rc][cq * 4 + 2] = s2;
  shm[rc][cq * 4 + 3] = m3; shs[rc][cq * 4 + 3] = s3;
  __syncthreads();
  if (rc == 0) {                    // threads 0..15, fixed-order merge over chunks
    for (int q = 1; q < 16; ++q) {
      lse_merge(m0, s0, shm[q][cq * 4 + 0], shs[q][cq * 4 + 0]);
      lse_merge(m1, s1, shm[q][cq * 4 + 1], shs[q][cq * 4 + 1]);
      lse_merge(m2, s2, shm[q][cq * 4 + 2], shs[q][cq * 4 + 2]);
      lse_merge(m3, s3, shm[q][cq * 4 + 3], shs[q][cq * 4 + 3]);
    }
    float4 out;
    out.x = EPSV * (log_mu() - (m0 + __logf(s0)));
    out.y = EPSV * (log_mu() - (m1 + __logf(s1)));
    out.z = EPSV * (log_mu() - (m2 + __logf(s2)));
    out.w = EPSV * (log_mu() - (m3 + __logf(s3)));
    ((float4*)(v + n * P))[qcol] = out;
  }
}

// ---------------- convergence check (reproduces while_loop cond) ----------------
__global__ void err_check(const float* __restrict__ errPartial, int* __restrict__ flag) {
  __shared__ float sh[256];
  if (*flag) return;
  float t = 0.0f;
  for (int i = threadIdx.x; i < NB * P / 8; i += 256) t += errPartial[i];
  sh[threadIdx.x] = t;
  __syncthreads();
  for (int off = 128; off; off >>= 1) {
    if (threadIdx.x < off) sh[threadIdx.x] += sh[threadIdx.x + off];
    __syncthreads();
  }
  if (threadIdx.x == 0) {
    float err = sh[0] / (float)NB;   // mean over batch of sum_i |du|
    if (err < THRESHV) *flag = 1;
  }
}

// ---------------- pi + per-row cost partials (b128 streaming) ----------------
__global__ void pi_cost(const float* __restrict__ C, const float* __restrict__ u,
                        const float* __restrict__ v, float* __restrict__ pi,
                        float* __restrict__ costPartial) {
  __shared__ float sh[8];
  int rg = blockIdx.x;
  int n  = rg >> 11;
  const float4* C4 = (const float4*)(C + (size_t)rg * P);
  float4*       p4 = (float4*)(pi + (size_t)rg * P);
  const float4* v4 = (const float4*)(v + n * P);
  float ui = u[rg];
  float acc = 0.0f;
  #pragma unroll
  for (int it = 0; it < P / 1024; ++it) {   // 2 iterations of float4 per thread
    int q = threadIdx.x + it * 256;
    float4 c = C4[q];
    float4 vv = v4[q];
    float4 pv;
    pv.x = __expf((ui + vv.x - c.x) * INV_EPS);
    pv.y = __expf((ui + vv.y - c.y) * INV_EPS);
    pv.z = __expf((ui + vv.z - c.z) * INV_EPS);
    pv.w = __expf((ui + vv.w - c.w) * INV_EPS);
    p4[q] = pv;
    acc += pv.x * c.x + pv.y * c.y + pv.z * c.z + pv.w * c.w;
  }
  #pragma unroll
  for (int off = 16; off; off >>= 1) acc += __shfl_xor(acc, off, 32);
  if ((threadIdx.x & 31) == 0) sh[threadIdx.x >> 5] = acc;
  __syncthreads();
  if (threadIdx.x == 0) {
    float t = 0.0f;
    #pragma unroll
    for (int i = 0; i < 8; ++i) t += sh[i];
    costPartial[rg] = t;
  }
}

// ---------------- deterministic final cost reduction ----------------
__global__ void cost_reduce(const float* __restrict__ costPartial, float* __restrict__ out) {
  __shared__ float sh[8];
  int n = blockIdx.x;
  const float4* cp4 = (const float4*)(costPartial + n * P);
  float acc = 0.0f;
  #pragma unroll
  for (int it = 0; it < P / 1024; ++it) {   // 2 iterations of float4 per thread
    float4 c = cp4[threadIdx.x + it * 256];
    acc += c.x + c.y + c.z + c.w;
  }
  #pragma unroll
  for (int off = 16; off; off >>= 1) acc += __shfl_xor(acc, off, 32);
  if ((threadIdx.x & 31) == 0) sh[threadIdx.x >> 5] = acc;
  __syncthreads();
  if (threadIdx.x == 0) {
    float t = 0.0f;
    #pragma unroll
    for (int i = 0; i < 8; ++i) t += sh[i];
    out[n] = t;
  }
}

extern "C" void kernel_launch(void* const* d_in, const int* in_sizes, int n_in,
                              void* d_out, int out_size, void* d_ws, size_t ws_size,
                              hipStream_t stream) {
  const float* x = (const float*)d_in[0];
  const float* y = (const float*)d_in[1];

  float* out  = (float*)d_out;
  float* cost = out;                               // [4]
  float* pi   = out + 4;                           // [4*2048*2048]
  float* Cm   = out + 4 + (size_t)NB * P * P;      // [4*2048*2048]

  float* ws          = (float*)d_ws;
  float* xs          = ws;                         // NB*P
  float* ys          = xs + NB * P;                // NB*P
  float* u           = ys + NB * P;                // NB*P
  float* v           = u + NB * P;                 // NB*P
  float* errPartial  = v + NB * P;                 // NB*P/8 = 1024
  float* costPartial = errPartial + NB * P / 8;    // NB*P
  int*   flag        = (int*)(costPartial + NB * P);

  // init u, v, flag
  init_uv<<<(NB * P + 255) / 256, 256, 0, stream>>>(u, v, flag);
  // row norms for x and y (wave per row, 8 waves/block)
  row_norms<<<(2 * NB * P) / 8, 256, 0, stream>>>(x, y, xs, ys);
  // C = |x|^2 + |y|^2 - 2 X Y^T  (WMMA, one wave per 16x16 tile)
  cost_wmma<<<(NB * 128 * 128) / 8, 256, 0, stream>>>(x, y, xs, ys, Cm);

  for (int it = 0; it < MAX_ITER; ++it) {
    u_update<<<NB * P / 8, 256, 0, stream>>>(Cm, v, u, errPartial, flag);
    v_update<<<NB * (P / 64), 256, 0, stream>>>(Cm, u, v, flag);
    err_check<<<1, 256, 0, stream>>>(errPartial, flag);
  }

  pi_cost<<<NB * P, 256, 0, stream>>>(Cm, u, v, pi, costPartial);
  cost_reduce<<<NB, 256, 0, stream>>>(costPartial, cost);
}